// NodeVarGraphConvolutionLayer_55508157333881
// MI455X (gfx1250) — compile-verified
//
#include <hip/hip_runtime.h>
#include <math.h>

// ---------------------------------------------------------------------------
// Problem constants (from reference)
// ---------------------------------------------------------------------------
static constexpr int Bsz = 8;       // batch
static constexpr int Nn  = 2048;    // nodes
static constexpr int Cin = 64;      // channels
static constexpr int NT  = 3;       // num_terms (A^1..A^3)
#define LN_EPS 1e-5f

#define BDIM 256                    // 8 waves per block (wave32)
#define KSPLIT 4                    // split K dim of GEMV across blocks

typedef float v2f __attribute__((ext_vector_type(2)));
typedef float v8f __attribute__((ext_vector_type(8)));

// wave32 butterfly sum (all lanes end with the full sum)
__device__ __forceinline__ float wave_sum(float v) {
#pragma unroll
    for (int m = 16; m >= 1; m >>= 1)
        v += __shfl_xor(v, m, 32);
    return v;
}

// ---------------------------------------------------------------------------
// Zero the accumulation buffers P[1..3] (atomics accumulate into them)
// ---------------------------------------------------------------------------
__global__ void zero_kernel(float* __restrict__ p, int n) {
    int i = blockIdx.x * blockDim.x + threadIdx.x;
    if (i < n) p[i] = 0.f;
}

// ---------------------------------------------------------------------------
// Stage 1: Xs[b,n] = sum_c X[b,n,c]   (one wave per node, 2 elems per lane)
// ---------------------------------------------------------------------------
__global__ void __launch_bounds__(BDIM)
reduce_x_kernel(const float* __restrict__ X, float* __restrict__ P0) {
    const int wave = threadIdx.x >> 5;
    const int lane = threadIdx.x & 31;
    const int node = blockIdx.x * (BDIM / 32) + wave;   // linear b*N + n
    const float* xp = X + (size_t)node * Cin;
    float v = xp[lane] + xp[lane + 32];
    v = wave_sum(v);
    if (lane == 0) P0[node] = v;
}

// ---------------------------------------------------------------------------
// Stage 2: batched GEMV  pout[b,:] += A[b] @ pin[b,:]  via V_WMMA_F32_16X16X4_F32
//   grid = (B, N/128, KSPLIT); 8 waves/block; each wave owns a 16-row strip.
//   A-operand (16x4 f32, 2 VGPR): lane L -> row m0+(L&15), k-pair k0+2*(L>=16)
//   B-operand: p[k] replicated over all 16 N-columns -> same per-lane k-pair
//   => every D column holds the partial dot product; lanes 0/16 write it out.
// ---------------------------------------------------------------------------
__global__ void __launch_bounds__(BDIM)
gemv_wmma_kernel(const float* __restrict__ A, const float* __restrict__ pin,
                 float* __restrict__ pout) {
    const int wave = threadIdx.x >> 5;
    const int lane = threadIdx.x & 31;
    const int b    = blockIdx.x;
    const int m0   = blockIdx.y * 128 + wave * 16;
    const int kbase = blockIdx.z * (Nn / KSPLIT);
    const int kend  = kbase + (Nn / KSPLIT);

    const int row   = m0 + (lane & 15);
    const int khalf = (lane >> 4) << 1;          // 0 or 2

    const float* __restrict__ Arow = A + ((size_t)b * Nn + row) * Nn;
    const float* __restrict__ pv   = pin + (size_t)b * Nn;

    v8f acc = {0.f, 0.f, 0.f, 0.f, 0.f, 0.f, 0.f, 0.f};

    for (int k0 = kbase; k0 < kend; k0 += 16) {
        // stream-ahead prefetch of the A row (global_prefetch_b8)
        __builtin_prefetch(Arow + k0 + 512, 0, 1);
#pragma unroll
        for (int kk = 0; kk < 16; kk += 4) {
            v2f a  = *(const v2f*)(Arow + k0 + kk + khalf);  // A[row, k..k+1]
            v2f bb = *(const v2f*)(pv   + k0 + kk + khalf);  // p[k..k+1] (replicated over N)
            acc = __builtin_amdgcn_wmma_f32_16x16x4_f32(
                /*neg_a=*/false, a, /*neg_b=*/false, bb,
                /*c_mod=*/(short)0, acc, /*reuse_a=*/false, /*reuse_b=*/false);
        }
    }

    // D layout: VGPR v, lanes 0-15 -> M=v, lanes 16-31 -> M=v+8; column = lane&15.
    // Column 0 holders are lane 0 (rows m0..m0+7) and lane 16 (rows m0+8..m0+15).
    if ((lane & 15) == 0) {
        float* outp = pout + (size_t)b * Nn + m0 + ((lane >> 4) << 3);
#pragma unroll
        for (int v = 0; v < 8; ++v)
            atomicAdd(&outp[v], acc[v]);
    }
}

// ---------------------------------------------------------------------------
// Stage 3: Y[b,n,c] = sum_i h[i,c,n] * P[i,b,n]; LayerNorm over c; tanh.
//   One wave per node; 2 channels per lane (c = lane, lane+32).
// ---------------------------------------------------------------------------
__global__ void __launch_bounds__(BDIM)
combine_ln_tanh_kernel(const float* __restrict__ h, const float* __restrict__ P,
                       const float* __restrict__ gamma, const float* __restrict__ beta,
                       float* __restrict__ out) {
    const int wave = threadIdx.x >> 5;
    const int lane = threadIdx.x & 31;
    const int node = blockIdx.x * (BDIM / 32) + wave;   // linear b*N + n
    const int n    = node & (Nn - 1);
    const int c0   = lane;
    const int c1   = lane + 32;

    float y0 = 0.f, y1 = 0.f;
#pragma unroll
    for (int i = 0; i <= NT; ++i) {
        const float p = P[(size_t)i * Bsz * Nn + node];
        y0 += h[((size_t)i * Cin + c0) * Nn + n] * p;
        y1 += h[((size_t)i * Cin + c1) * Nn + n] * p;
    }

    const float inv = 1.f / (float)Cin;
    const float mean = wave_sum(y0 + y1) * inv;
    const float e2   = wave_sum(y0 * y0 + y1 * y1) * inv;
    const float var  = e2 - mean * mean;
    const float rstd = rsqrtf(var + LN_EPS);

    float* op = out + (size_t)node * Cin;
    op[c0] = tanhf((y0 - mean) * rstd * gamma[c0] + beta[c0]);
    op[c1] = tanhf((y1 - mean) * rstd * gamma[c1] + beta[c1]);
}

// ---------------------------------------------------------------------------
// Launch
// ---------------------------------------------------------------------------
extern "C" void kernel_launch(void* const* d_in, const int* in_sizes, int n_in,
                              void* d_out, int out_size, void* d_ws, size_t ws_size,
                              hipStream_t stream) {
    const float* A     = (const float*)d_in[0];
    const float* X     = (const float*)d_in[1];
    const float* h     = (const float*)d_in[2];
    const float* gamma = (const float*)d_in[3];
    const float* beta  = (const float*)d_in[4];
    float* out = (float*)d_out;

    // workspace: P[4][B][N] fp32 = 256 KB
    float* P = (float*)d_ws;
    const size_t BN = (size_t)Bsz * Nn;

    // zero accumulation targets P[1..3]
    const int nz = 3 * Bsz * Nn;
    zero_kernel<<<nz / 256, 256, 0, stream>>>(P + BN, nz);

    // P[0] = Xs
    reduce_x_kernel<<<(Bsz * Nn) / (BDIM / 32), BDIM, 0, stream>>>(X, P);

    // P[t] = A @ P[t-1], t = 1..3
    dim3 g(Bsz, Nn / 128, KSPLIT);
    for (int t = 1; t <= NT; ++t)
        gemv_wmma_kernel<<<g, BDIM, 0, stream>>>(A, P + (size_t)(t - 1) * BN,
                                                 P + (size_t)t * BN);

    // combine + LN + tanh
    combine_ln_tanh_kernel<<<(Bsz * Nn) / (BDIM / 32), BDIM, 0, stream>>>(
        h, P, gamma, beta, out);
}